// PointnetSAModuleKNN_85066122265453
// MI455X (gfx1250) — compile-verified
//
#include <hip/hip_runtime.h>

// ---------------------------------------------------------------------------
// PointNet++ SA module (FPS + kNN + 3x [1x1 conv, BN(train), ReLU] + max-K)
// MI455X / gfx1250: fp32 WMMA (V_WMMA_F32_16X16X4_F32) for the conv GEMMs.
// ---------------------------------------------------------------------------

#define B_  8
#define N_  16384
#define S_  2048
#define K_  32
#define CIN 64
#define EPSF 1e-5f

constexpr int M_ = B_ * S_ * K_;          // 524288 grouped points
constexpr int MT_ = M_ / 16;              // 32768 column tiles

typedef float v2f __attribute__((ext_vector_type(2)));
typedef float v8f __attribute__((ext_vector_type(8)));

// ---------------------------------------------------------------------------
// 1) Farthest point sampling: one workgroup per batch, distance array lives
//    in registers (16 per thread), LDS argmax with first-index tie-break.
// ---------------------------------------------------------------------------
__global__ __launch_bounds__(1024)
void fps_kernel(const float* __restrict__ xyz, int* __restrict__ fps_idx)
{
    __shared__ float sval[1024];
    __shared__ int   sidx[1024];
    __shared__ int   sfar;

    const int b = blockIdx.x;
    const int t = threadIdx.x;
    const float* xb = xyz + (size_t)b * 3 * N_;   // (3, N) rows x,y,z

    float dist[16];
#pragma unroll
    for (int i = 0; i < 16; ++i) dist[i] = 1e10f;

    int far = 0;
    for (int s = 0; s < S_; ++s) {
        if (t == 0) fps_idx[b * S_ + s] = far;
        const float cx = xb[far], cy = xb[N_ + far], cz = xb[2 * N_ + far];

        float bestv = -1.0f; int besti = 0;
#pragma unroll
        for (int i = 0; i < 16; ++i) {
            const int n = t + (i << 10);           // strided, increasing
            const float dx = xb[n] - cx;
            const float dy = xb[N_ + n] - cy;
            const float dz = xb[2 * N_ + n] - cz;
            const float d  = dx * dx + dy * dy + dz * dz;
            const float dm = fminf(dist[i], d);
            dist[i] = dm;
            if (dm > bestv) { bestv = dm; besti = n; }   // strict > keeps lowest n
        }
        sval[t] = bestv; sidx[t] = besti;
        __syncthreads();
        for (int off = 512; off > 0; off >>= 1) {
            if (t < off) {
                const float v2 = sval[t + off]; const int i2 = sidx[t + off];
                if (v2 > sval[t] || (v2 == sval[t] && i2 < sidx[t])) {
                    sval[t] = v2; sidx[t] = i2;
                }
            }
            __syncthreads();
        }
        if (t == 0) sfar = sidx[0];
        __syncthreads();
        far = sfar;
    }
}

// ---------------------------------------------------------------------------
// 2) new_xyz output: (B, 3, S) gathered by fps_idx
// ---------------------------------------------------------------------------
__global__ void new_xyz_kernel(const float* __restrict__ xyz,
                               const int* __restrict__ fps_idx,
                               float* __restrict__ out_xyz)
{
    const int idx = blockIdx.x * blockDim.x + threadIdx.x;   // B*3*S
    if (idx >= B_ * 3 * S_) return;
    const int s  = idx % S_;
    const int bd = idx / S_;          // b*3 + d
    const int b  = bd / 3;
    const int f  = fps_idx[b * S_ + s];
    out_xyz[idx] = xyz[(size_t)bd * N_ + f];
}

// ---------------------------------------------------------------------------
// 3) kNN: one workgroup per centroid; full 16384-float distance row in LDS
//    (64 KB of the 320 KB WGP pool), then 32 argmin-extract rounds with
//    first-index tie-break (matches lax.top_k ordering).
// ---------------------------------------------------------------------------
__global__ __launch_bounds__(256)
void knn_kernel(const float* __restrict__ xyz,
                const int* __restrict__ fps_idx,
                int* __restrict__ knn_idx)
{
    __shared__ float dist[N_];
    __shared__ float rval[256];
    __shared__ int   ridx[256];

    const int cs = blockIdx.x;        // b*S + s
    const int b  = cs / S_;
    const int t  = threadIdx.x;
    const float* xb = xyz + (size_t)b * 3 * N_;

    const int f = fps_idx[cs];
    const float cx = xb[f], cy = xb[N_ + f], cz = xb[2 * N_ + f];

    for (int n = t; n < N_; n += 256) {
        const float dx = xb[n] - cx;
        const float dy = xb[N_ + n] - cy;
        const float dz = xb[2 * N_ + n] - cz;
        dist[n] = dx * dx + dy * dy + dz * dz;
    }
    __syncthreads();

    for (int k = 0; k < K_; ++k) {
        float bv = 3.4e38f; int bi = 0x7fffffff;
        for (int n = t; n < N_; n += 256) {
            const float d = dist[n];
            if (d < bv) { bv = d; bi = n; }          // increasing n: keeps lowest
        }
        rval[t] = bv; ridx[t] = bi;
        __syncthreads();
        for (int off = 128; off > 0; off >>= 1) {
            if (t < off) {
                const float v2 = rval[t + off]; const int i2 = ridx[t + off];
                if (v2 < rval[t] || (v2 == rval[t] && i2 < ridx[t])) {
                    rval[t] = v2; ridx[t] = i2;
                }
            }
            __syncthreads();
        }
        if (t == 0) { knn_idx[(size_t)cs * K_ + k] = ridx[0]; dist[ridx[0]] = 3.4e38f; }
        __syncthreads();
    }
}

// ---------------------------------------------------------------------------
// 4) Gather + concat -> X0 (cipad x M) row-major. Rows 0..2 centered xyz,
//    3..66 grouped point features, 67 zero pad (ci 67 -> 68).
// ---------------------------------------------------------------------------
__global__ void gather_concat(const float* __restrict__ xyz,
                              const float* __restrict__ points,
                              const int* __restrict__ fps_idx,
                              const int* __restrict__ knn_idx,
                              float* __restrict__ X0)
{
    const int m = blockIdx.x * blockDim.x + threadIdx.x;
    if (m >= M_) return;
    const int bs = m / K_;            // b*S + s
    const int b  = bs / S_;
    const int j  = knn_idx[m];
    const int f  = fps_idx[bs];

    const float* xb = xyz + (size_t)b * 3 * N_;
#pragma unroll
    for (int d = 0; d < 3; ++d)
        X0[(size_t)d * M_ + m] = xb[d * N_ + j] - xb[d * N_ + f];

    const float* pb = points + (size_t)b * CIN * N_;
#pragma unroll 8
    for (int c = 0; c < CIN; ++c)
        X0[(size_t)(3 + c) * M_ + m] = pb[(size_t)c * N_ + j];

    X0[(size_t)67 * M_ + m] = 0.0f;   // pad row
}

// ---------------------------------------------------------------------------
// 5) GEMM via V_WMMA_F32_16X16X4_F32: Y = W(CO x CI) * act(X(CI x M)).
//    One wave owns one 16-column m-tile and ALL CO/16 row tiles: the B
//    fragment is loaded once per k-step and reused across every co-tile,
//    cutting activation traffic to 1x. Weights are staged zero-padded in
//    LDS (<= 32 KB). act() = previous layer's BN+ReLU fused on load.
//    A layout (16x4): lanes 0-15 K=0/1 in v.x/v.y, lanes 16-31 K=2/3.
//    B layout (4x16): VGPR0 rows {0,2}, VGPR1 rows {1,3}, cols over 16 lanes.
//    C/D layout:      VGPR r -> rows r (lanes 0-15) and r+8 (lanes 16-31).
// ---------------------------------------------------------------------------
template <int CI, int CIPAD, int CO, bool BN>
__global__ __launch_bounds__(256)
void conv_wmma(const float* __restrict__ W, const float* __restrict__ X,
               const float* __restrict__ scale, const float* __restrict__ shift,
               float* __restrict__ Y)
{
    constexpr int COT = CO / 16;                 // co tiles per wave
    __shared__ float Wlds[CO * CIPAD];           // zero-padded weights

    // cooperative, padded weight stage
    for (int i = threadIdx.x; i < CO * CIPAD; i += 256) {
        const int r = i / CIPAD, c = i % CIPAD;
        Wlds[i] = (c < CI) ? W[r * CI + c] : 0.0f;
    }
    __syncthreads();

    const int lane = threadIdx.x & 31;
    const int wave = threadIdx.x >> 5;
    const int mt   = blockIdx.x * 8 + wave;      // exact: grid = MT_/8
    const int half = lane >> 4;                  // 0: lanes 0-15, 1: lanes 16-31
    const int l15  = lane & 15;
    const int col  = (mt << 4) + l15;

    v8f acc[COT];
#pragma unroll
    for (int ct = 0; ct < COT; ++ct) acc[ct] = v8f{};

#pragma unroll
    for (int kk = 0; kk < CIPAD; kk += 4) {
        const int k0 = kk + half * 2;
        const int k1 = k0 + 1;

        float b0 = X[(size_t)k0 * M_ + col];
        float b1 = X[(size_t)k1 * M_ + col];
        if (BN) {                                // fused BN+ReLU of prev layer
            b0 = fmaxf(b0 * scale[k0] + shift[k0], 0.0f);
            b1 = fmaxf(b1 * scale[k1] + shift[k1], 0.0f);
        }
        v2f bb; bb.x = b0; bb.y = b1;

#pragma unroll
        for (int ct = 0; ct < COT; ++ct) {
            v2f a;
            a.x = Wlds[(ct * 16 + l15) * CIPAD + k0];
            a.y = Wlds[(ct * 16 + l15) * CIPAD + k1];
            acc[ct] = __builtin_amdgcn_wmma_f32_16x16x4_f32(
                          false, a, false, bb, (short)0, acc[ct], false, false);
        }
    }

#pragma unroll
    for (int ct = 0; ct < COT; ++ct) {
        const int row0 = ct * 16 + half * 8;
#pragma unroll
        for (int r = 0; r < 8; ++r)
            Y[(size_t)(row0 + r) * M_ + col] = acc[ct][r];
    }
}

// ---------------------------------------------------------------------------
// 6) Training-mode BN statistics per channel -> folded scale/shift.
// ---------------------------------------------------------------------------
__global__ __launch_bounds__(256)
void bn_stats(const float* __restrict__ Y, const float* __restrict__ gamma,
              const float* __restrict__ beta, float* __restrict__ scale,
              float* __restrict__ shift)
{
    __shared__ float ss[256], sq[256];
    const int c = blockIdx.x;
    const int t = threadIdx.x;
    const float* row = Y + (size_t)c * M_;

    float s0 = 0.0f, s1 = 0.0f;
    for (int m = t; m < M_; m += 256) { const float v = row[m]; s0 += v; s1 += v * v; }
    ss[t] = s0; sq[t] = s1;
    __syncthreads();
    for (int off = 128; off > 0; off >>= 1) {
        if (t < off) { ss[t] += ss[t + off]; sq[t] += sq[t + off]; }
        __syncthreads();
    }
    if (t == 0) {
        const float mean = ss[0] / (float)M_;
        const float var  = sq[0] / (float)M_ - mean * mean;
        const float sc   = gamma[c] * rsqrtf(var + EPSF);
        scale[c] = sc;
        shift[c] = beta[c] - mean * sc;
    }
}

// ---------------------------------------------------------------------------
// 7) Apply BN2+ReLU and max over K=32 contiguous columns -> new_points.
// ---------------------------------------------------------------------------
__global__ void finalize_max(const float* __restrict__ Y2,
                             const float* __restrict__ scale,
                             const float* __restrict__ shift,
                             float* __restrict__ out_points)
{
    const int idx = blockIdx.x * blockDim.x + threadIdx.x;   // B*128*S
    if (idx >= B_ * 128 * S_) return;
    const int s  = idx % S_;
    const int bc = idx / S_;
    const int c  = bc & 127;
    const int b  = bc >> 7;

    const size_t mb = ((size_t)(b * S_ + s)) * K_;
    const float sc = scale[c], sh = shift[c];
    const float* row = Y2 + (size_t)c * M_ + mb;
    float best = 0.0f;                 // ReLU outputs are >= 0
#pragma unroll
    for (int k = 0; k < K_; ++k)
        best = fmaxf(best, fmaxf(row[k] * sc + sh, 0.0f));
    out_points[((size_t)b * 128 + c) * S_ + s] = best;
}

// ---------------------------------------------------------------------------
// Host-side orchestration
// ---------------------------------------------------------------------------
extern "C" void kernel_launch(void* const* d_in, const int* in_sizes, int n_in,
                              void* d_out, int out_size, void* d_ws, size_t ws_size,
                              hipStream_t stream)
{
    (void)in_sizes; (void)n_in; (void)out_size; (void)ws_size;

    const float* xyz    = (const float*)d_in[0];
    const float* points = (const float*)d_in[1];
    const float* W0 = (const float*)d_in[2];
    const float* g0 = (const float*)d_in[3];
    const float* b0 = (const float*)d_in[4];
    const float* W1 = (const float*)d_in[5];
    const float* g1 = (const float*)d_in[6];
    const float* b1 = (const float*)d_in[7];
    const float* W2 = (const float*)d_in[8];
    const float* g2 = (const float*)d_in[9];
    const float* b2 = (const float*)d_in[10];

    constexpr int CIP0 = 68;                 // 3 + 64 padded to mult of 4

    // Workspace layout (floats). Y2 (128*M) overlays X0+Y0 (132*M) after
    // both are dead (conv2 only reads Y1).
    float* ws = (float*)d_ws;
    float* X0 = ws;                              // 68*M
    float* Y0 = ws + (size_t)CIP0 * M_;          // 64*M
    float* Y2 = ws;                              // 128*M  (overlay)
    float* Y1 = ws + (size_t)(CIP0 + 64) * M_;   // 64*M
    float* tail    = ws + (size_t)(CIP0 + 64 + 64) * M_;
    int*   fps_idx = (int*)tail;                         // B*S
    int*   knn_idx = fps_idx + B_ * S_;                  // B*S*K
    float* scale0  = (float*)(knn_idx + M_);
    float* shift0  = scale0 + 64;
    float* scale1  = shift0 + 64;
    float* shift1  = scale1 + 64;
    float* scale2  = shift1 + 64;
    float* shift2  = scale2 + 128;

    float* out_xyz    = (float*)d_out;                   // B*3*S
    float* out_points = out_xyz + B_ * 3 * S_;           // B*128*S

    // 1) FPS (serial chain; one workgroup per batch)
    fps_kernel<<<B_, 1024, 0, stream>>>(xyz, fps_idx);

    // 2) new_xyz output
    new_xyz_kernel<<<(B_ * 3 * S_ + 255) / 256, 256, 0, stream>>>(xyz, fps_idx, out_xyz);

    // 3) kNN (one workgroup per centroid, 64 KB LDS distance row)
    knn_kernel<<<B_ * S_, 256, 0, stream>>>(xyz, fps_idx, knn_idx);

    // 4) Gather + concat -> X0 (68 x M)
    gather_concat<<<(M_ + 255) / 256, 256, 0, stream>>>(xyz, points, fps_idx, knn_idx, X0);

    // 5) Layer 0: Y0 = W0 (64x67) * X0 ; fp32 WMMA
    conv_wmma<67, CIP0, 64, false><<<MT_ / 8, 256, 0, stream>>>(W0, X0, nullptr, nullptr, Y0);
    bn_stats<<<64, 256, 0, stream>>>(Y0, g0, b0, scale0, shift0);

    // 6) Layer 1: Y1 = W1 (64x64) * BN0ReLU(Y0)
    conv_wmma<64, 64, 64, true><<<MT_ / 8, 256, 0, stream>>>(W1, Y0, scale0, shift0, Y1);
    bn_stats<<<64, 256, 0, stream>>>(Y1, g1, b1, scale1, shift1);

    // 7) Layer 2: Y2 = W2 (128x64) * BN1ReLU(Y1)   (Y2 overlays X0/Y0)
    conv_wmma<64, 64, 128, true><<<MT_ / 8, 256, 0, stream>>>(W2, Y1, scale1, shift1, Y2);
    bn_stats<<<128, 256, 0, stream>>>(Y2, g2, b2, scale2, shift2);

    // 8) BN2 + ReLU + max over K -> new_points
    finalize_max<<<(B_ * 128 * S_ + 255) / 256, 256, 0, stream>>>(Y2, scale2, shift2, out_points);
}